// SSGConv_27659589386301
// MI455X (gfx1250) — compile-verified
//
#include <hip/hip_runtime.h>
#include <stdint.h>

#define N_NODES 100000
#define N_EDGES 1600000
#define D 128
#define K_HOPS 5
#define ALPHA 0.1f
#define COEF ((1.0f - ALPHA) / (float)K_HOPS)
#define ND (N_NODES * D)

typedef __attribute__((ext_vector_type(2))) float v2f;
typedef __attribute__((ext_vector_type(8))) float v8f;

// ---------------- degree / normalization ----------------

__global__ void init_deg_kernel(float* __restrict__ deg) {
    int i = blockIdx.x * blockDim.x + threadIdx.x;
    if (i < N_NODES) deg[i] = 1.0f;  // self-loop contributes 1
}

__global__ void count_deg_kernel(const long long* __restrict__ ei,
                                 float* __restrict__ deg) {
    int e = blockIdx.x * blockDim.x + threadIdx.x;
    if (e < N_EDGES) {
        int col = (int)ei[(size_t)N_EDGES + e];
        __hip_atomic_fetch_add(&deg[col], 1.0f, __ATOMIC_RELAXED,
                               __HIP_MEMORY_SCOPE_AGENT);
    }
}

__global__ void dinv_kernel(float* __restrict__ deg) {
    int i = blockIdx.x * blockDim.x + threadIdx.x;
    if (i < N_NODES) deg[i] = rsqrtf(deg[i]);  // deg >= 1 always
}

// x_acc = alpha*x ; h0 = x ; h1 = dinv^2 * x  (self-loop prep for hop 0)
__global__ void init_vec_kernel(const float* __restrict__ x,
                                const float* __restrict__ dinv,
                                float* __restrict__ x_acc,
                                float* __restrict__ h0,
                                float* __restrict__ h1) {
    int idx = blockIdx.x * blockDim.x + threadIdx.x;
    if (idx < ND) {
        int node = idx >> 7;  // /D
        float v = x[idx];
        float di = dinv[node];
        x_acc[idx] = ALPHA * v;
        h0[idx] = v;
        h1[idx] = di * di * v;
    }
}

// ---------------- SpMM: one wave (32 lanes) per edge ----------------
// h_dst[row, :] += dinv[row]*dinv[col] * h_src[col, :]
__global__ __launch_bounds__(256) void spmm_edge_kernel(
        const long long* __restrict__ ei, const float* __restrict__ dinv,
        const float* __restrict__ h_src, float* __restrict__ h_dst) {
    int e = (int)((blockIdx.x * blockDim.x + threadIdx.x) >> 5);
    int lane = threadIdx.x & 31;
    if (e >= N_EDGES) return;
    int row = (int)ei[e];
    int col = (int)ei[(size_t)N_EDGES + e];
    float w = dinv[row] * dinv[col];
    float4 v = *(const float4*)(h_src + (size_t)col * D + lane * 4);
    float* dst = h_dst + (size_t)row * D + lane * 4;
    __hip_atomic_fetch_add(dst + 0, w * v.x, __ATOMIC_RELAXED, __HIP_MEMORY_SCOPE_AGENT);
    __hip_atomic_fetch_add(dst + 1, w * v.y, __ATOMIC_RELAXED, __HIP_MEMORY_SCOPE_AGENT);
    __hip_atomic_fetch_add(dst + 2, w * v.z, __ATOMIC_RELAXED, __HIP_MEMORY_SCOPE_AGENT);
    __hip_atomic_fetch_add(dst + 3, w * v.w, __ATOMIC_RELAXED, __HIP_MEMORY_SCOPE_AGENT);
}

// x_acc += COEF * h_new ; optionally h_prep = dinv^2 * h_new (next hop's
// self-loop initialized destination buffer)
template <bool PREP>
__global__ void post_spmm_kernel(const float* __restrict__ h_new,
                                 const float* __restrict__ dinv,
                                 float* __restrict__ x_acc,
                                 float* __restrict__ h_prep) {
    int idx = blockIdx.x * blockDim.x + threadIdx.x;
    if (idx < ND) {
        float v = h_new[idx];
        x_acc[idx] += COEF * v;
        if (PREP) {
            float di = dinv[idx >> 7];
            h_prep[idx] = di * di * v;
        }
    }
}

// ---------------- dense projection: out = x_acc @ W^T + bias ----------------
// V_WMMA_F32_16X16X4_F32, one wave per 16-row tile, K unrolled over 128.
// W (row-major [j][k]) staged in LDS with stride 132 floats -> conflict-free
// b64 B-fragment reads (banks 4n..4n+3 distinct across all 32 lanes).
#define LDS_STRIDE 132

__global__ __launch_bounds__(256) void gemm_wmma_kernel(
        const float* __restrict__ xa, const float* __restrict__ W,
        const float* __restrict__ bias, float* __restrict__ out) {
    __shared__ float ldsW[D * LDS_STRIDE];

    int t = threadIdx.x;
    // stage W: 4096 float4 loads across 256 threads
    for (int i = t; i < D * (D / 4); i += 256) {
        int r = i >> 5;        // row j
        int c4 = i & 31;       // float4 within row
        float4 v = ((const float4*)W)[r * 32 + c4];
        *(float4*)&ldsW[r * LDS_STRIDE + c4 * 4] = v;
    }
    __syncthreads();

    int wave = (int)((blockIdx.x * blockDim.x + t) >> 5);
    int m0 = wave * 16;
    if (m0 >= N_NODES) return;

    int lane = t & 31;
    int n = lane & 15;       // A row / B col / C col
    int khalf = lane >> 4;   // selects K pair {2*khalf, 2*khalf+1}

    const float* arow = xa + (size_t)(m0 + n) * D;  // A: row n of the tile

    for (int jt = 0; jt < 8; ++jt) {
        v8f c = {};
        float bb = bias[jt * 16 + n];
        const float* brow = &ldsW[(jt * 16 + n) * LDS_STRIDE];  // B col n = W row
        #pragma unroll 8
        for (int k = 0; k < D; k += 4) {
            int kb = k + khalf * 2;
            v2f a = *(const v2f*)(arow + kb);
            v2f b = *(const v2f*)(brow + kb);
            c = __builtin_amdgcn_wmma_f32_16x16x4_f32(
                    false, a, false, b, (short)0, c, false, false);
        }
        #pragma unroll
        for (int v = 0; v < 8; ++v) {
            int rr = v + khalf * 8;  // C/D: VGPR v holds row v (+8 for hi half)
            out[(size_t)(m0 + rr) * D + jt * 16 + n] = c[v] + bb;
        }
    }
}

// ---------------- host orchestration ----------------

extern "C" void kernel_launch(void* const* d_in, const int* in_sizes, int n_in,
                              void* d_out, int out_size, void* d_ws, size_t ws_size,
                              hipStream_t stream) {
    const float* x = (const float*)d_in[0];
    const long long* ei = (const long long*)d_in[1];  // int64 per reference
    const float* W = (const float*)d_in[2];
    const float* bias = (const float*)d_in[3];
    float* out = (float*)d_out;

    // workspace layout (256B aligned)
    char* ws = (char*)d_ws;
    size_t off = 0;
    float* deg = (float*)(ws + off);          off += ((size_t)N_NODES * 4 + 255) & ~(size_t)255;
    float* h0 = (float*)(ws + off);           off += (size_t)ND * 4;
    float* h1 = (float*)(ws + off);           off += (size_t)ND * 4;
    float* x_acc = (float*)(ws + off);        off += (size_t)ND * 4;
    (void)ws_size;

    const int TB = 256;
    int gN = (N_NODES + TB - 1) / TB;
    int gE = (N_EDGES + TB - 1) / TB;
    int gND = (ND + TB - 1) / TB;
    int gSPMM = (N_EDGES * 32 + TB - 1) / TB;  // one wave per edge

    init_deg_kernel<<<gN, TB, 0, stream>>>(deg);
    count_deg_kernel<<<gE, TB, 0, stream>>>(ei, deg);
    dinv_kernel<<<gN, TB, 0, stream>>>(deg);  // deg now holds d^-1/2
    init_vec_kernel<<<gND, TB, 0, stream>>>(x, deg, x_acc, h0, h1);

    float* hA = h0;  // source
    float* hB = h1;  // destination (pre-initialized with self-loop term)
    for (int k = 0; k < K_HOPS; ++k) {
        spmm_edge_kernel<<<gSPMM, TB, 0, stream>>>(ei, deg, hA, hB);
        if (k < K_HOPS - 1) {
            post_spmm_kernel<true><<<gND, TB, 0, stream>>>(hB, deg, x_acc, hA);
        } else {
            post_spmm_kernel<false><<<gND, TB, 0, stream>>>(hB, deg, x_acc, nullptr);
        }
        float* tmp = hA; hA = hB; hB = tmp;
    }

    // 6250 row tiles, 8 waves (tiles) per 256-thread block
    int tiles = N_NODES / 16;
    int gGEMM = (tiles + 7) / 8;
    gemm_wmma_kernel<<<gGEMM, TB, 0, stream>>>(x_acc, W, bias, out);
}